// HarmonicOscillator_2310692406020
// MI455X (gfx1250) — compile-verified
//
#include <hip/hip_runtime.h>
#include <math.h>

// ---------------- constants matching the reference ----------------
#define NHARM 64
#define TFRAMES 1025
#define BATCH 4
#define TOUT 262144              // (1025-1)*256
#define NCHUNK 1024              // TOUT / 256
#define OMEGA_F ((float)(2.0 * 3.14159265358979323846 / 44100.0))
#define RF      ((float)(1024.0 / 262143.0))        // (t-1)/(T-1), align_corners ratio
#define PI_MASK 3.14159274f                          // float32(np.pi)
#define TWO_PI   6.28318530717958647692f
#define INV_2PI  0.15915494309189533577f

typedef float v2f __attribute__((ext_vector_type(2)));
typedef float v8f __attribute__((ext_vector_type(8)));

// ---------------- kernel 1: frame-rate fs / amps ----------------
// grid = BATCH*TFRAMES blocks, block = 64 threads (one per harmonic)
__global__ void ho_prep_kernel(const float* __restrict__ f0,
                               const float* __restrict__ master,
                               const float* __restrict__ ov,
                               float* __restrict__ fsF,
                               float* __restrict__ ampF) {
    const int bt = blockIdx.x;
    const int b  = bt / TFRAMES;
    const int ti = bt % TFRAMES;
    const int o  = threadIdx.x;

    const float f  = f0[b * TFRAMES + ti];
    const float fs = f * (float)(o + 1) * OMEGA_F;
    const float a  = ov[((size_t)(b * NHARM + o)) * TFRAMES + ti];
    const float ma = (fs > PI_MASK) ? 0.0f : a;

    __shared__ float red[64];
    red[o] = ma;
    __syncthreads();
    for (int s = 32; s > 0; s >>= 1) {
        if (o < s) red[o] += red[o + s];
        __syncthreads();
    }
    const float sum = red[0];
    const float amp = (ma / sum) * master[b * TFRAMES + ti];

    const size_t g = ((size_t)(b * NHARM + o)) * TFRAMES + ti;
    fsF[g]  = fs;
    ampF[g] = amp;
}

// ---------------- kernel 2: 256-sample chunk sums + exclusive scan ----------------
// grid = BATCH*NHARM blocks, block = 1024 threads (one per chunk)
__global__ void ho_scan_kernel(const float* __restrict__ fsF,
                               const float* __restrict__ initu,
                               float* __restrict__ chunkPhase) {
    const int bo = blockIdx.x;                 // (b*64 + o)
    const int ci = threadIdx.x;                // chunk index 0..1023
    const float* fsb = fsF + (size_t)bo * TFRAMES;

    float s = 0.0f;
    const int jbase = ci * 256;
    for (int k = 0; k < 256; ++k) {
        const int j = jbase + k;
        const float pos = (float)j * RF;
        int i0 = (int)floorf(pos);
        if (i0 > TFRAMES - 2) i0 = TFRAMES - 2;
        const float w = pos - (float)i0;
        float v = fsb[i0] * (1.0f - w) + fsb[i0 + 1] * w;
        if (j == 0) {
            const float u = initu[bo];
            v = 3.14159265f * (u * 2.0f - 1.0f);   // random initial phase
        }
        s += v;
    }

    __shared__ float buf[2][1024];
    int pin = 0;
    buf[0][ci] = s;
    __syncthreads();
    for (int off = 1; off < 1024; off <<= 1) {
        float v = buf[pin][ci];
        if (ci >= off) v += buf[pin][ci - off];
        buf[pin ^ 1][ci] = v;
        __syncthreads();
        pin ^= 1;
    }
    const float incl = buf[pin][ci];
    float excl = incl - s;                                  // phase before chunk start
    excl -= TWO_PI * rintf(excl * INV_2PI);                 // wrap (sin is 2pi-periodic)
    chunkPhase[(size_t)bo * NCHUNK + ci] = excl;
}

// ---------------- kernel 3: WMMA matmul-scan + sin synthesis ----------------
// grid = (NCHUNK, BATCH), block = 32 threads (one wave; EXEC all-ones for WMMA)
__global__ __launch_bounds__(32)
void ho_synth_kernel(const float* __restrict__ fsF,
                     const float* __restrict__ ampF,
                     const float* __restrict__ chunkPhase,
                     const float* __restrict__ initu,
                     float* __restrict__ out) {
    const int ci   = blockIdx.x;
    const int b    = blockIdx.y;
    const int lane = threadIdx.x;
    const int n    = lane & 15;       // harmonic column within a 16-group
    const int half = lane >> 4;       // lane half (selects K/M subranges per ISA layouts)
    const int m    = lane & 15;       // A/D row index within half

    const int j0c = ci * 256;
    float posA = (float)j0c * RF;
    int iA = (int)floorf(posA);
    if (iA > TFRAMES - 2) iA = TFRAMES - 2;

    // Stage the (at most) 3 frames this chunk touches, all 64 harmonics, into LDS.
    __shared__ float sFs[3 * 64];
    __shared__ float sAmp[3 * 64];
    for (int idx = lane; idx < 192; idx += 32) {
        const int fr = idx >> 6;
        const int o  = idx & 63;
        int gi = iA + fr;
        if (gi > TFRAMES - 1) gi = TFRAMES - 1;
        const size_t g = ((size_t)(b * NHARM + o)) * TFRAMES + gi;
        sFs[idx]  = fsF[g];
        sAmp[idx] = ampF[g];
    }
    __syncthreads();

    float carry[4];
    float initv[4];
#pragma unroll
    for (int og = 0; og < 4; ++og) {
        const int o = og * 16 + n;
        carry[og] = chunkPhase[(size_t)(b * NHARM + o) * NCHUNK + ci];
        const float u = initu[b * NHARM + o];
        initv[og] = 3.14159265f * (u * 2.0f - 1.0f);
    }

    // Sample 0 (j==0) lives only in chunk 0, tile 0, lane-half 0, WMMA slot c=0,h=0.
    const bool sel0base = (ci == 0) && (half == 0);

    float* outb = out + (size_t)b * TOUT;

    for (int tt = 0; tt < 16; ++tt) {
        const int j0 = j0c + tt * 16;
        const bool tile0 = sel0base && (tt == 0);

        // ---- hoisted interp params (shared across all 4 harmonic groups) ----
        // B-matrix slots: e = 2*c+h -> tile sample s = 4*c + 2*half + h
        int   frB[8]; float wB[8], oB[8];
#pragma unroll
        for (int e = 0; e < 8; ++e) {
            const int s = 4 * (e >> 1) + 2 * half + (e & 1);
            const float pos = (float)(j0 + s) * RF;
            int i0 = (int)floorf(pos);
            if (i0 > TFRAMES - 2) i0 = TFRAMES - 2;
            const float w = pos - (float)i0;
            int fr = i0 - iA;
            fr = fr < 0 ? 0 : (fr > 1 ? 1 : fr);
            frB[e] = fr * 64;
            wB[e]  = w;
            oB[e]  = 1.0f - w;
        }
        // D rows (amp interp): sample m_r = r + 8*half
        int   frD[8]; float wD[8], oD[8];
#pragma unroll
        for (int r = 0; r < 8; ++r) {
            const float pos = (float)(j0 + r + 8 * half) * RF;
            int i0 = (int)floorf(pos);
            if (i0 > TFRAMES - 2) i0 = TFRAMES - 2;
            const float w = pos - (float)i0;
            int fr = i0 - iA;
            fr = fr < 0 ? 0 : (fr > 1 ? 1 : fr);
            frD[r] = fr * 64;
            wD[r]  = w;
            oD[r]  = 1.0f - w;
        }

        float acc[8];
#pragma unroll
        for (int r = 0; r < 8; ++r) acc[r] = 0.0f;

#pragma unroll
        for (int og = 0; og < 4; ++og) {
            const int o = og * 16 + n;

            // D = L(16x16 lower-triangular ones) * X(16 samples x 16 harmonics)
            // as 4 chained V_WMMA_F32_16X16X4_F32 -> inclusive phase scan per column.
            v8f d = {0.f, 0.f, 0.f, 0.f, 0.f, 0.f, 0.f, 0.f};
#pragma unroll
            for (int c = 0; c < 4; ++c) {
                const int klo = 4 * c + 2 * half;
                v2f a;
                a.x = (klo     <= m) ? 1.0f : 0.0f;   // L[m, klo]
                a.y = (klo + 1 <= m) ? 1.0f : 0.0f;   // L[m, klo+1]
                v2f bb;
                {
                    const int e = 2 * c;
                    float v = sFs[frB[e] + o] * oB[e] + sFs[frB[e] + 64 + o] * wB[e];
                    if (c == 0) v = tile0 ? initv[og] : v;   // init phase at j==0 only
                    bb.x = v;
                }
                {
                    const int e = 2 * c + 1;
                    bb.y = sFs[frB[e] + o] * oB[e] + sFs[frB[e] + 64 + o] * wB[e];
                }
                d = __builtin_amdgcn_wmma_f32_16x16x4_f32(
                        false, a, false, bb, (short)0, d, false, false);
            }

            // phase -> hardware sin -> amp-weight -> accumulate over harmonics
#pragma unroll
            for (int r = 0; r < 8; ++r) {
                const float ph  = d[r] + carry[og];
                const float sv  = __sinf(ph);   // carry wrapped => |ph| small, v_sin safe
                const float amp = sAmp[frD[r] + o] * oD[r] + sAmp[frD[r] + 64 + o] * wD[r];
                acc[r] += sv * amp;
            }

            // carry += D[15, n] (lives in VGPR7 of lane 16+n); keep it wrapped.
            const float d15 = __shfl(d[7], 16 + n, 32);
            float cn = carry[og] + d15;
            cn -= TWO_PI * rintf(cn * INV_2PI);
            carry[og] = cn;
        }

        // Reduce over the 16 harmonic lanes of each half (xor<16 stays in-half).
#pragma unroll
        for (int r = 0; r < 8; ++r) {
            float v = acc[r];
            v += __shfl_xor(v, 1, 32);
            v += __shfl_xor(v, 2, 32);
            v += __shfl_xor(v, 4, 32);
            v += __shfl_xor(v, 8, 32);
            acc[r] = v;
        }
        if (lane == 0) {
            *(float4*)(outb + j0)     = make_float4(acc[0], acc[1], acc[2], acc[3]);
            *(float4*)(outb + j0 + 4) = make_float4(acc[4], acc[5], acc[6], acc[7]);
        }
        if (lane == 16) {
            *(float4*)(outb + j0 + 8)  = make_float4(acc[0], acc[1], acc[2], acc[3]);
            *(float4*)(outb + j0 + 12) = make_float4(acc[4], acc[5], acc[6], acc[7]);
        }
    }
}

// ---------------- launcher ----------------
extern "C" void kernel_launch(void* const* d_in, const int* in_sizes, int n_in,
                              void* d_out, int out_size, void* d_ws, size_t ws_size,
                              hipStream_t stream) {
    const float* f0     = (const float*)d_in[0];   // [4,1,1025]
    const float* master = (const float*)d_in[1];   // [4,1,1025]
    const float* ov     = (const float*)d_in[2];   // [4,1,64,1025]
    const float* initu  = (const float*)d_in[3];   // [4,1,64]
    float* out = (float*)d_out;                    // [4,1,262144]

    float* fsF        = (float*)d_ws;                          // 4*64*1025
    float* ampF       = fsF  + (size_t)BATCH * NHARM * TFRAMES;
    float* chunkPhase = ampF + (size_t)BATCH * NHARM * TFRAMES; // 4*64*1024

    ho_prep_kernel<<<BATCH * TFRAMES, 64, 0, stream>>>(f0, master, ov, fsF, ampF);
    ho_scan_kernel<<<BATCH * NHARM, 1024, 0, stream>>>(fsF, initu, chunkPhase);
    dim3 grid(NCHUNK, BATCH);
    ho_synth_kernel<<<grid, 32, 0, stream>>>(fsF, ampF, chunkPhase, initu, out);
}